// EBEMLoss_5677946765570
// MI455X (gfx1250) — compile-verified
//
#include <hip/hip_runtime.h>
#include <hip/hip_bf16.h>
#include <stdint.h>

// Problem constants (fixed by the reference): x, P are (N, C) fp32.
#define NROWS 65536
#define NCOLS 1000
#define ROWS_PER_BLOCK 8          // one wave32 per row, 8 waves per block
#define BLOCKS (NROWS / ROWS_PER_BLOCK)
#define F4_PER_ROW (NCOLS / 4)    // 250 float4 per row
#define F4_ITERS 8                // ceil(250/32)

typedef __attribute__((ext_vector_type(4))) unsigned int uint32x4_t_;
typedef __attribute__((ext_vector_type(8))) int          int32x8_t_;
typedef __attribute__((ext_vector_type(4))) int          int32x4_t_;

// Issue one TDM 2D tile load: ROWS_PER_BLOCK rows x NCOLS fp32 from global -> LDS.
// D# layout per CDNA5 ISA ch.8 (group0 128b, group1 256b; groups 2/3 zero for 2D).
// clang-23 lane: 6-arg builtin (uint32x4, int32x8, int32x4, int32x4, int32x8, i32 cpol).
__device__ __forceinline__ void tdm_load_tile(unsigned lds_byte_off, const float* gptr) {
    unsigned long long ga = (unsigned long long)(uintptr_t)gptr;
    uint32x4_t_ g0;
    g0[0] = 1u;                                            // count=1, is_restore=0, gather off
    g0[1] = lds_byte_off;                                  // lds_addr (bytes)
    g0[2] = (unsigned)ga;                                  // global_addr[31:0]
    g0[3] = ((unsigned)(ga >> 32) & 0x01FFFFFFu)           // global_addr[56:32]
          | (2u << 30);                                    // type = 2 ("image")
    int32x8_t_ g1;
    g1[0] = (int)(2u << 16);                               // workgroup_mask=0, data_size=2 (4B)
    g1[1] = (int)((unsigned)NCOLS << 16);                  // tensor_dim0[15:0] in [31:16]
    g1[2] = (int)((unsigned)ROWS_PER_BLOCK << 16);         // tensor_dim0 hi=0, tensor_dim1 lo
    g1[3] = (int)((unsigned)NCOLS << 16);                  // tensor_dim1 hi=0, tile_dim0=NCOLS
    g1[4] = (int)ROWS_PER_BLOCK;                           // tile_dim1=8, tile_dim2=0
    g1[5] = (int)NCOLS;                                    // tensor_dim0_stride low 32 (elements)
    g1[6] = 0;                                             // stride hi, tensor_dim1_stride lo
    g1[7] = 0;
    int32x4_t_ gz4; gz4[0] = 0; gz4[1] = 0; gz4[2] = 0; gz4[3] = 0;
    int32x8_t_ gz8;
    #pragma unroll
    for (int i = 0; i < 8; ++i) gz8[i] = 0;
    __builtin_amdgcn_tensor_load_to_lds(g0, g1, gz4, gz4, gz8, 0);
}

__device__ __forceinline__ float wave_sum32(float v) {
    #pragma unroll
    for (int off = 16; off >= 1; off >>= 1) v += __shfl_xor(v, off, 32);
    return v;
}
__device__ __forceinline__ float wave_max32(float v) {
    #pragma unroll
    for (int off = 16; off >= 1; off >>= 1) v = fmaxf(v, __shfl_xor(v, off, 32));
    return v;
}

// Kernel 1: per-row  rowval = (m + log(sum exp(x-m))) * sum(P) - dot(P, x)
__global__ __launch_bounds__(256) void ebem_rows_kernel(
        const float* __restrict__ x, const float* __restrict__ p,
        float* __restrict__ rowout, int use_atomic, float* __restrict__ out) {
    __shared__ float xt[ROWS_PER_BLOCK * NCOLS];
    __shared__ float pt[ROWS_PER_BLOCK * NCOLS];

    const int row0 = blockIdx.x * ROWS_PER_BLOCK;
    const int wave = threadIdx.x >> 5;
    const int lane = threadIdx.x & 31;

    // Wave 0 drives the Tensor Data Mover for both tiles (EXEC-independent op,
    // issued once per executing wave), then waits its private TENSORcnt.
    if (threadIdx.x < 32) {
        tdm_load_tile((unsigned)(uintptr_t)&xt[0], x + (size_t)row0 * NCOLS);
        tdm_load_tile((unsigned)(uintptr_t)&pt[0], p + (size_t)row0 * NCOLS);
        __builtin_amdgcn_s_wait_tensorcnt(0);
    }
    __syncthreads();

    const float4* xrow = (const float4*)&xt[wave * NCOLS];
    const float4* prow = (const float4*)&pt[wave * NCOLS];

    // Pass 1: row max (keep x in registers: 8 x float4 = 32 VGPRs/lane).
    float4 xs[F4_ITERS];
    float m = -INFINITY;
    #pragma unroll
    for (int t = 0; t < F4_ITERS; ++t) {
        int j = lane + 32 * t;
        if (j < F4_PER_ROW) {
            xs[t] = xrow[j];
            m = fmaxf(m, fmaxf(fmaxf(xs[t].x, xs[t].y), fmaxf(xs[t].z, xs[t].w)));
        }
    }
    m = wave_max32(m);

    // Pass 2: exp-sum, sum(P), dot(P, x).
    float s = 0.0f, sp = 0.0f, dt = 0.0f;
    #pragma unroll
    for (int t = 0; t < F4_ITERS; ++t) {
        int j = lane + 32 * t;
        if (j < F4_PER_ROW) {
            float4 pv = prow[j];
            float4 xv = xs[t];
            s += __expf(xv.x - m) + __expf(xv.y - m) +
                 __expf(xv.z - m) + __expf(xv.w - m);
            sp += (pv.x + pv.y) + (pv.z + pv.w);
            dt = fmaf(pv.x, xv.x, fmaf(pv.y, xv.y, fmaf(pv.z, xv.z, fmaf(pv.w, xv.w, dt))));
        }
    }
    s  = wave_sum32(s);
    sp = wave_sum32(sp);
    dt = wave_sum32(dt);

    if (lane == 0) {
        float rowval = (m + __logf(s)) * sp - dt;
        if (use_atomic) {
            atomicAdd(out, rowval * (1.0f / (float)NROWS));
        } else {
            rowout[row0 + wave] = rowval;
        }
    }
}

__device__ __forceinline__ float block256_sum(float v) {
    __shared__ float sm[8];
    int tid = threadIdx.x;
    v = wave_sum32(v);
    if ((tid & 31) == 0) sm[tid >> 5] = v;
    __syncthreads();
    float t = 0.0f;
    if (tid < 8) {
        t = sm[tid];
        #pragma unroll
        for (int off = 4; off >= 1; off >>= 1) t += __shfl_xor(t, off, 8);
    }
    return t;   // valid in tid==0
}

// Kernel 2a: 65536 rowvals -> 256 partials (deterministic tree).
__global__ __launch_bounds__(256) void ebem_reduce_partial(
        const float* __restrict__ in, float* __restrict__ out) {
    float v = in[blockIdx.x * 256 + threadIdx.x];
    float t = block256_sum(v);
    if (threadIdx.x == 0) out[blockIdx.x] = t;
}

// Kernel 2b: 256 partials -> scalar loss (scaled by 1/N).
__global__ __launch_bounds__(256) void ebem_reduce_final(
        const float* __restrict__ in, float* __restrict__ out) {
    float v = in[threadIdx.x];
    float t = block256_sum(v);
    if (threadIdx.x == 0) out[0] = t * (1.0f / (float)NROWS);
}

__global__ void ebem_zero_out(float* out) { out[0] = 0.0f; }

extern "C" void kernel_launch(void* const* d_in, const int* in_sizes, int n_in,
                              void* d_out, int out_size, void* d_ws, size_t ws_size,
                              hipStream_t stream) {
    const float* x = (const float*)d_in[0];
    const float* p = (const float*)d_in[1];
    float* out = (float*)d_out;
    float* ws  = (float*)d_ws;

    const size_t need = (size_t)(NROWS + 256) * sizeof(float);
    if (ws_size >= need) {
        ebem_rows_kernel<<<BLOCKS, 256, 0, stream>>>(x, p, ws, 0, out);
        ebem_reduce_partial<<<NROWS / 256, 256, 0, stream>>>(ws, ws + NROWS);
        ebem_reduce_final<<<1, 256, 0, stream>>>(ws + NROWS, out);
    } else {
        ebem_zero_out<<<1, 1, 0, stream>>>(out);
        ebem_rows_kernel<<<BLOCKS, 256, 0, stream>>>(x, p, nullptr, 1, out);
    }
}